// RGATLayer_68917045231753
// MI455X (gfx1250) — compile-verified
//
#include <hip/hip_runtime.h>
#include <hip/hip_bf16.h>

// ---- problem constants (match reference) ----
#define N_NODES 50000
#define N_EDGES 800000
#define IN_DIM  128
#define OUT_DIM 64
#define HEADS   4
#define NUM_RELS 20
#define NUM_BASES 8
#define HD (HEADS * OUT_DIM)          // 256
#define TWO_D (2 * OUT_DIM)           // 128

typedef float v2f __attribute__((ext_vector_type(2)));
typedef float v8f __attribute__((ext_vector_type(8)));

// ---------------------------------------------------------------------------
// Kernel 1: basis combination  attn_fcs[r, c, h] = sum_b w_comp[r,b] * aw[b,c,h]
// 20 * 512 = 10240 outputs; trivially parallel.
// ---------------------------------------------------------------------------
__global__ void rgat_basis_kernel(const float* __restrict__ wcomp,
                                  const float* __restrict__ aw,
                                  float* __restrict__ attn_fcs) {
    int idx = blockIdx.x * blockDim.x + threadIdx.x;
    if (idx >= NUM_RELS * TWO_D * HEADS) return;
    int r   = idx >> 9;          // / 512
    int rem = idx & 511;         // c*4 + h
    float s = 0.f;
#pragma unroll
    for (int b = 0; b < NUM_BASES; ++b)
        s += wcomp[r * NUM_BASES + b] * aw[b * (TWO_D * HEADS) + rem];
    attn_fcs[idx] = s;
}

// ---------------------------------------------------------------------------
// Kernel 2: fp32 WMMA GEMM   C[M x ncols] = A[M x 128] @ W[ncols x 128]^T
// One wave (32 threads) per 16x16 output tile -> EXEC all ones for WMMA.
// Uses V_WMMA_F32_16X16X4_F32, K-loop of 32 steps over IN_DIM=128.
//
// A-frag (16x4 f32): lane<16 holds row M=lane, K=k0,k0+1; lane>=16 holds K=k0+2,k0+3.
// B-frag (4x16):     lane%16 = column N, same K split; B[k][n] = W[n][k].
// C/D (16x16 f32):   VGPR j -> M = j + 8*(lane>=16), N = lane%16.
// ---------------------------------------------------------------------------
__global__ void rgat_wmma_gemm_f32(const float* __restrict__ A,
                                   const float* __restrict__ W,
                                   float* __restrict__ C,
                                   int ncols) {
    const int m0   = blockIdx.x * 16;
    const int n0   = blockIdx.y * 16;
    const int lane = threadIdx.x;       // 0..31
    const int half = lane >> 4;         // 0 or 1
    const int lr   = lane & 15;

    const float* arow = A + (size_t)(m0 + lr) * IN_DIM + half * 2;
    const float* wrow = W + (size_t)(n0 + lr) * IN_DIM + half * 2;

    v8f acc = {};
#pragma unroll 4
    for (int k = 0; k < IN_DIM; k += 4) {
        v2f a = *(const v2f*)(arow + k);
        v2f b = *(const v2f*)(wrow + k);
        // D = A(16x4) * B(4x16) + C ; fp32 throughout
        acc = __builtin_amdgcn_wmma_f32_16x16x4_f32(
                  /*neg_a=*/false, a, /*neg_b=*/false, b,
                  /*c_mod=*/(short)0, acc, /*reuse_a=*/false, /*reuse_b=*/false);
    }

#pragma unroll
    for (int j = 0; j < 8; ++j) {
        int m = m0 + half * 8 + j;
        C[(size_t)m * ncols + n0 + lr] = acc[j];
    }
}

// ---------------------------------------------------------------------------
// Kernel 3: edge kernel — one wave32 per edge.
//   Lane l holds z2[c] for c = {l, l+32, l+64, l+96}  (z2 = [z[src], z[dst]])
//   4 head dot-products reduced across the wave with shfl_xor,
//   leaky_relu, then 8 atomic f32 adds per lane into out[dst].
// ---------------------------------------------------------------------------
__global__ void rgat_edge_kernel(const float* __restrict__ z,
                                 const float* __restrict__ attn_fcs,
                                 const int* __restrict__ src,
                                 const int* __restrict__ dst,
                                 const int* __restrict__ rel,
                                 float* __restrict__ out) {
    const int lane = threadIdx.x & 31;
    const int e = (blockIdx.x * blockDim.x + threadIdx.x) >> 5;  // wave id
    if (e >= N_EDGES) return;   // uniform per wave

    const int s = src[e];
    const int d = dst[e];
    const int r = rel[e];

    const float* zs = z + (size_t)s * OUT_DIM;
    const float* zd = z + (size_t)d * OUT_DIM;

    // z2 slices held by this lane
    const float zs0 = zs[lane];        // c = lane        (src, d = lane)
    const float zs1 = zs[lane + 32];   // c = lane + 32   (src, d = lane+32)
    const float zd0 = zd[lane];        // c = lane + 64
    const float zd1 = zd[lane + 32];   // c = lane + 96

    // relation weight rows for this lane's 4 c-indices (each row = 4 heads, contiguous)
    const float* w = attn_fcs + (size_t)r * (TWO_D * HEADS);
    const float4 w0 = *(const float4*)(w + (size_t)(lane)      * HEADS);
    const float4 w1 = *(const float4*)(w + (size_t)(lane + 32) * HEADS);
    const float4 w2 = *(const float4*)(w + (size_t)(lane + 64) * HEADS);
    const float4 w3 = *(const float4*)(w + (size_t)(lane + 96) * HEADS);

    float acc[HEADS];
    acc[0] = zs0 * w0.x + zs1 * w1.x + zd0 * w2.x + zd1 * w3.x;
    acc[1] = zs0 * w0.y + zs1 * w1.y + zd0 * w2.y + zd1 * w3.y;
    acc[2] = zs0 * w0.z + zs1 * w1.z + zd0 * w2.z + zd1 * w3.z;
    acc[3] = zs0 * w0.w + zs1 * w1.w + zd0 * w2.w + zd1 * w3.w;

    // wave32 xor-reduction: all lanes end with the full sum
#pragma unroll
    for (int h = 0; h < HEADS; ++h) {
        float v = acc[h];
        for (int off = 16; off > 0; off >>= 1)
            v += __shfl_xor(v, off, 32);
        // leaky_relu(0.01)
        acc[h] = v > 0.f ? v : 0.01f * v;
    }

    // message scatter: out[d, h*64 + col] += a[h] * zsrc[col], col in {lane, lane+32}
    float* o = out + (size_t)d * HD;
#pragma unroll
    for (int h = 0; h < HEADS; ++h) {
        unsafeAtomicAdd(&o[h * OUT_DIM + lane],      acc[h] * zs0);
        unsafeAtomicAdd(&o[h * OUT_DIM + lane + 32], acc[h] * zs1);
    }
}

// ---------------------------------------------------------------------------
// launch
// ---------------------------------------------------------------------------
extern "C" void kernel_launch(void* const* d_in, const int* in_sizes, int n_in,
                              void* d_out, int out_size, void* d_ws, size_t ws_size,
                              hipStream_t stream) {
    const float* h          = (const float*)d_in[0];
    const float* fc_w       = (const float*)d_in[1];
    const float* self_fc_w  = (const float*)d_in[2];
    const float* aw         = (const float*)d_in[3];
    const float* wcomp      = (const float*)d_in[4];
    const int*   src        = (const int*)d_in[5];
    const int*   dst        = (const int*)d_in[6];
    const int*   rel        = (const int*)d_in[7];
    float*       out        = (float*)d_out;

    // workspace: z (N x 64 f32) then attn_fcs (20 x 128 x 4 f32)
    float* z    = (float*)d_ws;
    float* attn = z + (size_t)N_NODES * OUT_DIM;

    // 1) basis combination -> attn_fcs
    rgat_basis_kernel<<<(NUM_RELS * TWO_D * HEADS + 255) / 256, 256, 0, stream>>>(
        wcomp, aw, attn);

    // 2) z = h @ fc_w^T   (50000 x 64), WMMA tiles 16x16, one wave per tile
    rgat_wmma_gemm_f32<<<dim3(N_NODES / 16, OUT_DIM / 16), 32, 0, stream>>>(
        h, fc_w, z, OUT_DIM);

    // 3) out = h @ self_fc_w^T  (50000 x 256)  -- full overwrite initializes d_out
    rgat_wmma_gemm_f32<<<dim3(N_NODES / 16, HD / 16), 32, 0, stream>>>(
        h, self_fc_w, out, HD);

    // 4) edge attention + message scatter-add (one wave per edge, 8 waves/block)
    rgat_edge_kernel<<<N_EDGES / 8, 256, 0, stream>>>(z, attn, src, dst, rel, out);
}